// RSAGE_50800873177232
// MI455X (gfx1250) — compile-verified
//
#include <hip/hip_runtime.h>

typedef float v2f __attribute__((ext_vector_type(2)));
typedef float v8f __attribute__((ext_vector_type(8)));

#define DFEAT 128   // IN_FEAT == HID == 128
#define AP 132      // padded LDS row stride for A (conflict-free b64 frag reads)

// ---------------------------------------------------------------------------
// Utility: zero a float region (grid-stride, capture-safe replacement for memset)
// ---------------------------------------------------------------------------
__global__ void zero_f32(float* __restrict__ p, long long n) {
    long long i = (long long)blockIdx.x * blockDim.x + threadIdx.x;
    long long stride = (long long)gridDim.x * blockDim.x;
    for (; i < n; i += stride) p[i] = 0.0f;
}

// ---------------------------------------------------------------------------
// In-degree count per destination node (float, matches segment_sum of ones)
// ---------------------------------------------------------------------------
__global__ void deg_count(const int* __restrict__ dst, float* __restrict__ deg, int E) {
    int e = blockIdx.x * blockDim.x + threadIdx.x;
    if (e < E) atomicAdd(&deg[dst[e]], 1.0f);
}

// ---------------------------------------------------------------------------
// Edge-weighted scatter-ADD: agg[dst, f] += x[src, f] * ew   (one thread per
// (edge, feature); 128 consecutive threads share an edge -> coalesced x reads
// and conflict-free coalesced atomics)
// ---------------------------------------------------------------------------
__global__ void scatter_add(const float* __restrict__ x,
                            const int* __restrict__ src, const int* __restrict__ dst,
                            const float* __restrict__ ew,
                            float* __restrict__ agg, long long total) {
    long long gid = (long long)blockIdx.x * blockDim.x + threadIdx.x;
    if (gid >= total) return;
    int e = (int)(gid >> 7);
    int f = (int)(gid & 127);
    float v = x[(size_t)src[e] * DFEAT + f] * ew[e];
    atomicAdd(&agg[(size_t)dst[e] * DFEAT + f], v);
}

// ---------------------------------------------------------------------------
// Edge-weighted scatter-MAX. Messages are relu(hp)*ew >= 0, and agg is zero-
// initialized, so IEEE float max == unsigned-int max on the bit patterns.
// Zero init also yields 0 for isolated nodes (matches reference where()).
// ---------------------------------------------------------------------------
__global__ void scatter_max(const float* __restrict__ hp,
                            const int* __restrict__ src, const int* __restrict__ dst,
                            const float* __restrict__ ew,
                            float* __restrict__ agg, long long total) {
    long long gid = (long long)blockIdx.x * blockDim.x + threadIdx.x;
    if (gid >= total) return;
    int e = (int)(gid >> 7);
    int f = (int)(gid & 127);
    float v = hp[(size_t)src[e] * DFEAT + f] * ew[e];
    atomicMax((unsigned int*)&agg[(size_t)dst[e] * DFEAT + f], __float_as_uint(v));
}

// ---------------------------------------------------------------------------
// Mean normalization: agg[i,f] /= max(deg[i], 1)
// ---------------------------------------------------------------------------
__global__ void normalize_mean(float* __restrict__ agg, const float* __restrict__ deg,
                               long long total) {
    long long gid = (long long)blockIdx.x * blockDim.x + threadIdx.x;
    if (gid >= total) return;
    agg[gid] /= fmaxf(deg[gid >> 7], 1.0f);
}

// ---------------------------------------------------------------------------
// Fused multi-source WMMA GEMM:
//   C[M,KOUT] (= or +=) sum_p A_p[M,128] @ W_p[128,KOUT]  (+ bias0 + bias1) (relu?)
//
// One workgroup = 4 wave32s = 64 output rows; each wave owns a 16-row stripe
// and KOUT/16 16x16 f32 accumulators driven by V_WMMA_F32_16X16X4_F32.
//
// Data movement:
//  * 64x128 A block staged coalesced (float4) into LDS, row stride AP=132
//    -> each lane's A fragment is one contiguous conflict-free ds_load_b64.
//  * W staged in 32-row K-slabs using a K-PAIR INTERLEAVED layout:
//      element (k, n) -> sW[(k/2)*WSTR + 2n + (k&1)],  WSTR = 2*KOUT + 32
//    so a lane's B fragment (W[k][n], W[k+1][n]) is one contiguous
//    ds_load_b64 (no register-shuffle movs), and WSTR % 64 == 32 makes the
//    two K-halves of the wave hit disjoint bank sets (conflict-free).
//
// Fragment layouts (CDNA5 ISA 7.12.2):
//   A 16x4 f32 : lane -> M = lane%16, vgpr v -> K = 2*(lane/16) + v
//   B 4x16 f32 : lane -> N = lane%16, vgpr v -> K = 2*(lane/16) + v
//   C 16x16    : vgpr j -> M = j + 8*(lane/16), lane -> N = lane%16
//
// No early return: WMMA needs EXEC all-ones and LDS staging needs all waves
// at the barriers. Out-of-range stripes clamp their staging rows and skip
// only the epilogue store (nrows = 100000 is 16-aligned -> wave-uniform).
// ---------------------------------------------------------------------------
template <int KOUT>
__global__ __launch_bounds__(128) void gemm_fused(
    const float* __restrict__ A0, const float* __restrict__ W0,
    const float* __restrict__ A1, const float* __restrict__ W1,
    const float* __restrict__ A2, const float* __restrict__ W2,
    const float* __restrict__ A3, const float* __restrict__ W3,
    int npairs,
    const float* __restrict__ bias0, const float* __restrict__ bias1,
    float* __restrict__ C, int nrows, int accumulate, int do_relu) {
    constexpr int NT = KOUT / 16;
    constexpr int WSTR = 2 * KOUT + 32;          // interleaved K-pair row stride

    __shared__ float sA[4 * 16 * AP];            // 64 padded A rows (33.8 KB)
    __shared__ float sW[16 * WSTR];              // one 32-row W slab, interleaved

    const int tid  = threadIdx.x;
    const int wave = tid >> 5;
    const int lane = tid & 31;
    const int lo = lane & 15;                    // M for A, N for B/C
    const int khalf = lane >> 4;                 // K-half selector

    const int rowBlock = blockIdx.x * 64;        // first row of this workgroup
    const int rowBase  = rowBlock + wave * 16;
    const bool active  = (rowBase < nrows);

    v8f acc[NT];
    const v8f vzero = {0.f, 0.f, 0.f, 0.f, 0.f, 0.f, 0.f, 0.f};
#pragma unroll
    for (int t = 0; t < NT; ++t) acc[t] = vzero;

    float* __restrict__ sAw = sA + wave * 16 * AP;   // this wave's A stripe

    auto do_pair = [&](const float* __restrict__ A, const float* __restrict__ W) {
        __syncthreads();   // previous compute (or pair) done with sA/sW
        // ---- stage 64x128 A block, coalesced float4, rows clamped ----
        for (int i = tid; i < (64 * DFEAT) / 4; i += 128) {
            int linear = i * 4;
            int r = linear / DFEAT;
            int c = linear % DFEAT;
            int grow = rowBlock + r;
            if (grow >= nrows) grow = nrows - 1;
            float4 v = *(const float4*)(A + (size_t)grow * DFEAT + c);
            *(float4*)&sA[r * AP + c] = v;
        }
        // ---- K slabs of 32 rows (16 K-pairs) ----
        for (int ks = 0; ks < 4; ++ks) {
            __syncthreads();   // compute of previous slab done with sW
            // stage slab into K-pair interleaved layout:
            // unit = (kp, col-pair): load float2 from rows 2kp and 2kp+1,
            // store interleaved b128 {r0.x, r1.x, r0.y, r1.y}
            for (int u = tid; u < 16 * (KOUT / 2); u += 128) {
                int kp = u / (KOUT / 2);                  // slab-local K pair
                int n  = (u % (KOUT / 2)) * 2;            // column pair base
                int r0 = ks * 32 + 2 * kp;
                v2f g0 = *(const v2f*)(W + (size_t)r0 * KOUT + n);
                v2f g1 = *(const v2f*)(W + (size_t)(r0 + 1) * KOUT + n);
                float4 q;
                q.x = g0.x; q.y = g1.x; q.z = g0.y; q.w = g1.y;
                *(float4*)&sW[kp * WSTR + 2 * n] = q;
            }
            __syncthreads();   // sA (first slab) and sW staged
#pragma unroll
            for (int kk = 0; kk < 32; kk += 4) {
                const int kb = ks * 32 + kk + 2 * khalf;  // global K for A
                const int kp = kk / 2 + khalf;            // slab-local K pair
                const v2f a = *(const v2f*)&sAw[lo * AP + kb];
#pragma unroll
                for (int t = 0; t < NT; ++t) {
                    const v2f b = *(const v2f*)&sW[kp * WSTR + 2 * (t * 16 + lo)];
                    acc[t] = __builtin_amdgcn_wmma_f32_16x16x4_f32(
                        false, a, false, b, (short)0, acc[t], false, false);
                }
            }
        }
    };

    if (npairs > 0) do_pair(A0, W0);
    if (npairs > 1) do_pair(A1, W1);
    if (npairs > 2) do_pair(A2, W2);
    if (npairs > 3) do_pair(A3, W3);

    // ---- epilogue: optional accumulate with old C, bias (col-only), relu ----
    if (active) {
#pragma unroll
        for (int t = 0; t < NT; ++t) {
            float bv = 0.0f;
            if (bias0) bv += bias0[t * 16 + lo];
            if (bias1) bv += bias1[t * 16 + lo];
#pragma unroll
            for (int j = 0; j < 8; ++j) {
                const int row = rowBase + j + 8 * khalf;
                const size_t idx = (size_t)row * KOUT + t * 16 + lo;
                float v = acc[t][j] + bv;
                if (accumulate) v += C[idx];
                if (do_relu) v = fmaxf(v, 0.0f);
                C[idx] = v;
            }
        }
    }
}

// ---------------------------------------------------------------------------
// Host-side launcher
// ---------------------------------------------------------------------------
extern "C" void kernel_launch(void* const* d_in, const int* in_sizes, int n_in,
                              void* d_out, int out_size, void* d_ws, size_t ws_size,
                              hipStream_t stream) {
    // ---- inputs, in setup_inputs() dict order ----
    const float* x    = (const float*)d_in[0];
    const int*   src0 = (const int*)d_in[1];
    const int*   dst0 = (const int*)d_in[2];
    const float* ew0  = (const float*)d_in[3];
    const int*   src1 = (const int*)d_in[4];
    const int*   dst1 = (const int*)d_in[5];
    const float* ew1  = (const float*)d_in[6];
    const float* Ws0_0 = (const float*)d_in[7];
    const float* Wn0_0 = (const float*)d_in[8];
    const float* b0_0  = (const float*)d_in[9];
    const float* Wp_0  = (const float*)d_in[10];
    const float* bp_0  = (const float*)d_in[11];
    const float* Ws1_0 = (const float*)d_in[12];
    const float* Wn1_0 = (const float*)d_in[13];
    const float* b1_0  = (const float*)d_in[14];
    const float* Ws0_1 = (const float*)d_in[15];
    const float* Wn0_1 = (const float*)d_in[16];
    const float* b0_1  = (const float*)d_in[17];
    const float* Wp_1  = (const float*)d_in[18];
    const float* bp_1  = (const float*)d_in[19];
    const float* Ws1_1 = (const float*)d_in[20];
    const float* Wn1_1 = (const float*)d_in[21];
    const float* b1_1  = (const float*)d_in[22];

    const int N  = in_sizes[0] / DFEAT;   // 100000
    const int E0 = in_sizes[1];           // 600000
    const int E1 = in_sizes[4];

    // ---- workspace layout (floats): deg0 | deg1 | bufA | bufB | h ----
    float* ws   = (float*)d_ws;
    float* deg0 = ws;
    float* deg1 = deg0 + N;
    float* bufA = deg1 + N;                       // agg_r0 sum/mean -> hp
    float* bufB = bufA + (size_t)N * DFEAT;       // agg_r1 sum/mean -> maxagg
    float* h    = bufB + (size_t)N * DFEAT;       // hidden layer output
    float* out  = (float*)d_out;                  // [N, 64]

    const long long nf   = (long long)N * DFEAT;
    const long long tE0  = (long long)E0 * DFEAT;
    const long long tE1  = (long long)E1 * DFEAT;
    const int ZB = 256, ZG = 4096;
    const int SB = 256;
    const int gemmGrid = (N + 63) / 64;           // 64 rows per workgroup

    // 1) zero deg0,deg1,bufA,bufB (contiguous)
    zero_f32<<<ZG, ZB, 0, stream>>>(deg0, 2LL * N + 2LL * nf);

    // 2) in-degrees
    deg_count<<<(E0 + SB - 1) / SB, SB, 0, stream>>>(dst0, deg0, E0);
    deg_count<<<(E1 + SB - 1) / SB, SB, 0, stream>>>(dst1, deg1, E1);

    // 3) weighted scatter-add per relation
    scatter_add<<<(int)((tE0 + SB - 1) / SB), SB, 0, stream>>>(x, src0, dst0, ew0, bufA, tE0);
    scatter_add<<<(int)((tE1 + SB - 1) / SB), SB, 0, stream>>>(x, src1, dst1, ew1, bufB, tE1);

    // 4) mean normalize
    normalize_mean<<<(int)((nf + SB - 1) / SB), SB, 0, stream>>>(bufA, deg0, nf);
    normalize_mean<<<(int)((nf + SB - 1) / SB), SB, 0, stream>>>(bufB, deg1, nf);

    // 5) layer 0 (fused 4-source GEMM + both biases + relu):
    //    h = relu(x@Ws0_r0 + agg0@Wn0_r0 + x@Ws0_r1 + agg1@Wn0_r1 + b0_r0 + b0_r1)
    gemm_fused<128><<<gemmGrid, 128, 0, stream>>>(
        x, Ws0_0, bufA, Wn0_0, x, Ws0_1, bufB, Wn0_1, 4,
        b0_0, b0_1, h, N, /*accumulate=*/0, /*relu=*/1);

    // ---- relation r0 classifier path ----
    // 6) hp0 = relu(h @ Wp_r0 + bp_r0)  -> bufA (agg0 is dead)
    gemm_fused<128><<<gemmGrid, 128, 0, stream>>>(
        h, Wp_0, nullptr, nullptr, nullptr, nullptr, nullptr, nullptr, 1,
        bp_0, nullptr, bufA, N, 0, 1);
    // 7) maxagg0 buffer = 0
    zero_f32<<<ZG, ZB, 0, stream>>>(bufB, nf);
    // 8) maxagg0 = segment_max(hp0[src]*ew)
    scatter_max<<<(int)((tE0 + SB - 1) / SB), SB, 0, stream>>>(bufA, src0, dst0, ew0, bufB, tE0);
    // 9) logits = h@Ws1_r0 + maxagg0@Wn1_r0 + b1_r0 + b1_r1
    gemm_fused<64><<<gemmGrid, 128, 0, stream>>>(
        h, Ws1_0, bufB, Wn1_0, nullptr, nullptr, nullptr, nullptr, 2,
        b1_0, b1_1, out, N, /*accumulate=*/0, /*relu=*/0);

    // ---- relation r1 classifier path ----
    // 10) hp1 = relu(h @ Wp_r1 + bp_r1) -> bufA
    gemm_fused<128><<<gemmGrid, 128, 0, stream>>>(
        h, Wp_1, nullptr, nullptr, nullptr, nullptr, nullptr, nullptr, 1,
        bp_1, nullptr, bufA, N, 0, 1);
    // 11) maxagg1 buffer = 0
    zero_f32<<<ZG, ZB, 0, stream>>>(bufB, nf);
    // 12) maxagg1 = segment_max(hp1[src]*ew)
    scatter_max<<<(int)((tE1 + SB - 1) / SB), SB, 0, stream>>>(bufA, src1, dst1, ew1, bufB, tE1);
    // 13) logits += h@Ws1_r1 + maxagg1@Wn1_r1
    gemm_fused<64><<<gemmGrid, 128, 0, stream>>>(
        h, Ws1_1, bufB, Wn1_1, nullptr, nullptr, nullptr, nullptr, 2,
        nullptr, nullptr, out, N, /*accumulate=*/1, /*relu=*/0);
}